// FastSRModel_52158082842758
// MI455X (gfx1250) — compile-verified
//
#include <hip/hip_runtime.h>

// FastSRModel on MI455X (gfx1250, wave32).
//
// Stage 1: O(N^2) pairwise smooth-cutoff RBF features.
//   - All 4096 positions staged in LDS (48 KB << 320 KB/WGP).
//   - Branch on d^2 < cutoff^2 so only in-cutoff pairs (few %) pay for
//     sqrt/cos/exp transcendentals.
//   - The 16 Gaussian exponents -(u-k*D)^2/(2*eta^2) form a quadratic in k:
//     evaluated with constant-second-difference recurrence in log2 domain,
//     one v_exp_f32 per RBF center.
// Stage 2: f32 MLP via V_WMMA_F32_16X16X4_F32.
//   - One wave owns 16 atoms (rows). Layer1: K=16 -> 4 k-steps x 4 N-tiles.
//     Layer2: K=64 -> 16 k-steps x 4 N-tiles. Activations staged in LDS to
//     convert D-matrix layout -> A-matrix layout between layers.
//   - Layer3 (64 -> 1) + global energy sum: per-lane dot products, wave32
//     shuffle reduction, one atomicAdd per wave.

#define N_ATOMS  4096
#define N_RBF    16
#define N_HID    64
#define CUTOFF   5.0f

typedef __attribute__((ext_vector_type(2))) float v2f;
typedef __attribute__((ext_vector_type(8))) float v8f;

__device__ __forceinline__ v8f wmma4_f32(v2f a, v2f b, v8f c) {
  // 8 args: (neg_a, A, neg_b, B, c_mod, C, reuse_a, reuse_b)
  return __builtin_amdgcn_wmma_f32_16x16x4_f32(false, a, false, b, (short)0, c,
                                               false, false);
}

// ---------------------------------------------------------------------------
// Stage 1: per-atom RBF-aggregated features (N_ATOMS x N_RBF, f32).
// ---------------------------------------------------------------------------
__global__ __launch_bounds__(256) void rbf_features_kernel(
    const float* __restrict__ pos, float* __restrict__ feats) {
  __shared__ float sx[N_ATOMS];
  __shared__ float sy[N_ATOMS];
  __shared__ float sz[N_ATOMS];

  const int tid = threadIdx.x;
  for (int j = tid; j < N_ATOMS; j += 256) {
    sx[j] = pos[3 * j + 0];
    sy[j] = pos[3 * j + 1];
    sz[j] = pos[3 * j + 2];
  }
  __syncthreads();

  const int i = blockIdx.x * 256 + tid;
  const float xi = sx[i], yi = sy[i], zi = sz[i];

  float f[N_RBF];
#pragma unroll
  for (int k = 0; k < N_RBF; ++k) f[k] = 0.0f;

  constexpr float kEta   = 0.5f * CUTOFF / (float)N_RBF;     // 0.15625
  constexpr float kInv   = 1.0f / (2.0f * kEta * kEta);      // 20.48
  constexpr float kLog2e = 1.4426950408889634f;
  constexpr float K2     = kInv * kLog2e;                    // log2-domain scale
  constexpr float kDelta = (CUTOFF - 0.5f) / (float)(N_RBF - 1);  // 0.3
  constexpr float kDD    = -2.0f * kDelta * kDelta * K2;     // 2nd difference
  constexpr float kPi    = 3.14159265358979323846f;
  constexpr float kInvCut = 1.0f / CUTOFF;

  for (int j = 0; j < N_ATOMS; ++j) {
    const float dx = xi - sx[j];
    const float dy = yi - sy[j];
    const float dz = zi - sz[j];
    const float d2 = dx * dx + dy * dy + dz * dz;
    if (d2 < CUTOFF * CUTOFF && d2 > 1e-12f) {
      const float d = __fsqrt_rn(d2);
      const float w = 0.5f * (1.0f + __cosf(kPi * (d * kInvCut)));
      const float u = d - 0.5f;  // distance to first center
      float arg   = -u * u * K2;                                 // log2 gauss_0
      float delta = (2.0f * kDelta * u - kDelta * kDelta) * K2;  // 1st diff
#pragma unroll
      for (int k = 0; k < N_RBF; ++k) {
        f[k] = fmaf(w, exp2f(arg), f[k]);
        arg += delta;
        delta += kDD;
      }
    }
  }

#pragma unroll
  for (int k = 0; k < N_RBF; ++k) feats[i * N_RBF + k] = f[k];
}

// ---------------------------------------------------------------------------
// Stage 2: MLP (16 -> 64 -> 64 -> 1) + total-energy reduction, f32 WMMA.
// One wave32 handles a 16-atom slab. 4 waves per block -> 64 atoms/block.
// ---------------------------------------------------------------------------
__device__ __forceinline__ float silu(float v) {
  return v / (1.0f + __expf(-v));
}

__global__ __launch_bounds__(128) void mlp_energy_kernel(
    const float* __restrict__ feats,
    const float* __restrict__ W1, const float* __restrict__ b1,
    const float* __restrict__ W2, const float* __restrict__ b2,
    const float* __restrict__ W3, const float* __restrict__ b3,
    float* __restrict__ out) {
  __shared__ float h1s[4][16 * N_HID];
  __shared__ float h2s[4][16 * N_HID];

  const int lane = threadIdx.x & 31;
  const int wave = threadIdx.x >> 5;
  const int half = lane >> 4;   // selects K pair in A/B fragments
  const int col  = lane & 15;   // M index (A) / N index (B, C)
  const int m0   = (blockIdx.x * 4 + wave) * 16;

  float* __restrict__ h1 = h1s[wave];
  float* __restrict__ h2 = h2s[wave];

  // ---- Layer 1: (16 x 16) @ (16 x 64) --------------------------------------
  v8f acc[4];
#pragma unroll
  for (int n = 0; n < 4; ++n) acc[n] = 0.0f;

#pragma unroll
  for (int kc = 0; kc < N_RBF; kc += 4) {
    v2f a;
    a[0] = feats[(m0 + col) * N_RBF + kc + 2 * half + 0];
    a[1] = feats[(m0 + col) * N_RBF + kc + 2 * half + 1];
#pragma unroll
    for (int n = 0; n < 4; ++n) {
      v2f b;
      b[0] = W1[(kc + 2 * half + 0) * N_HID + n * 16 + col];
      b[1] = W1[(kc + 2 * half + 1) * N_HID + n * 16 + col];
      acc[n] = wmma4_f32(a, b, acc[n]);
    }
  }

#pragma unroll
  for (int n = 0; n < 4; ++n) {
#pragma unroll
    for (int r = 0; r < 8; ++r) {
      const int m = r + 8 * half;
      const int c = n * 16 + col;
      h1[m * N_HID + c] = silu(acc[n][r] + b1[c]);
    }
  }
  __syncthreads();

  // ---- Layer 2: (16 x 64) @ (64 x 64) --------------------------------------
#pragma unroll
  for (int n = 0; n < 4; ++n) acc[n] = 0.0f;

#pragma unroll
  for (int kc = 0; kc < N_HID; kc += 4) {
    v2f a;
    a[0] = h1[col * N_HID + kc + 2 * half + 0];
    a[1] = h1[col * N_HID + kc + 2 * half + 1];
#pragma unroll
    for (int n = 0; n < 4; ++n) {
      v2f b;
      b[0] = W2[(kc + 2 * half + 0) * N_HID + n * 16 + col];
      b[1] = W2[(kc + 2 * half + 1) * N_HID + n * 16 + col];
      acc[n] = wmma4_f32(a, b, acc[n]);
    }
  }

#pragma unroll
  for (int n = 0; n < 4; ++n) {
#pragma unroll
    for (int r = 0; r < 8; ++r) {
      const int m = r + 8 * half;
      const int c = n * 16 + col;
      h2[m * N_HID + c] = silu(acc[n][r] + b2[c]);
    }
  }
  __syncthreads();

  // ---- Layer 3 + energy reduction ------------------------------------------
  float p = 0.0f;
  for (int idx = lane; idx < 16 * N_HID; idx += 32) {
    p = fmaf(h2[idx], W3[idx & (N_HID - 1)], p);
  }
  if (lane == 0) p += 16.0f * b3[0];  // per-row bias for this 16-atom slab

#pragma unroll
  for (int off = 16; off > 0; off >>= 1) p += __shfl_down(p, off, 32);
  if (lane == 0) atomicAdd(out, p);
}

// ---------------------------------------------------------------------------
extern "C" void kernel_launch(void* const* d_in, const int* in_sizes, int n_in,
                              void* d_out, int out_size, void* d_ws,
                              size_t ws_size, hipStream_t stream) {
  const float* pos = (const float*)d_in[0];
  const float* W1  = (const float*)d_in[1];
  const float* b1  = (const float*)d_in[2];
  const float* W2  = (const float*)d_in[3];
  const float* b2  = (const float*)d_in[4];
  const float* W3  = (const float*)d_in[5];
  const float* b3  = (const float*)d_in[6];
  float* out   = (float*)d_out;
  float* feats = (float*)d_ws;  // N_ATOMS x N_RBF f32 = 256 KB scratch

  hipMemsetAsync(out, 0, sizeof(float), stream);

  rbf_features_kernel<<<N_ATOMS / 256, 256, 0, stream>>>(pos, feats);

  // 4 waves/block, 16 atoms per wave -> 64 atoms per block.
  mlp_energy_kernel<<<N_ATOMS / 64, 128, 0, stream>>>(feats, W1, b1, W2, b2,
                                                      W3, b3, out);
}